// GCN_19705309954252
// MI455X (gfx1250) — compile-verified
//
#include <hip/hip_runtime.h>
#include <hip/hip_bf16.h>
#include <stdint.h>

// ---------------------------------------------------------------------------
// GCN (3x GCNConv, 128->128->128->128) for MI455X / gfx1250.
//
//  - Dense GEMMs: FP32 WMMA (v_wmma_f32_16x16x4_f32), A panel staged to LDS
//    with CDNA5 async LDS fill (global_load_async_to_lds_b128 + s_wait_asynccnt).
//  - Aggregation: edges counting-sorted by dst once (600K int atomics + small
//    scan) -> pull-mode gather per node, NO fp atomics in steady state.
//    Bias + LeakyReLU fused into the aggregation store.
//  Steady-state per layer ~ 51MB GEMM + ~307MB random gather + 25.6MB store
//  => HBM-bound at ~roofline; atomic bottleneck eliminated.
// ---------------------------------------------------------------------------

typedef __attribute__((ext_vector_type(2))) float v2f;
typedef __attribute__((ext_vector_type(8))) float v8f;

#define LEAKY_SLOPE 0.01f

__device__ __forceinline__ v8f wmma_f32_16x16x4(v2f a, v2f b, v8f c) {
    // 8 args: (neg_a, A, neg_b, B, c_mod, C, reuse_a, reuse_b)
    return __builtin_amdgcn_wmma_f32_16x16x4_f32(
        false, a, false, b, (short)0, c, false, false);
}

// ------------------------- normalization precompute -------------------------

__global__ void k_clear(float* __restrict__ rowsum, float* __restrict__ deg,
                        int* __restrict__ cnt, int N) {
    int i = blockIdx.x * blockDim.x + threadIdx.x;
    if (i < N) { rowsum[i] = 0.0f; deg[i] = 0.0f; cnt[i] = 0; }
}

__global__ void k_rowsum(const int* __restrict__ src, const float* __restrict__ ew,
                         float* __restrict__ rowsum, int E) {
    int e = blockIdx.x * blockDim.x + threadIdx.x;
    if (e < E) unsafeAtomicAdd(&rowsum[src[e]], ew[e]);
}

__global__ void k_rinv(const float* __restrict__ rowsum, float* __restrict__ rinv, int N) {
    int i = blockIdx.x * blockDim.x + threadIdx.x;
    if (i < N) { float r = rowsum[i]; rinv[i] = (r > 0.0f) ? (1.0f / r) : 0.0f; }
}

__global__ void k_what_deg(const int* __restrict__ src, const int* __restrict__ dst,
                           const float* __restrict__ ew, const float* __restrict__ rinv,
                           float* __restrict__ w_hat, float* __restrict__ deg, int E) {
    int e = blockIdx.x * blockDim.x + threadIdx.x;
    if (e < E) {
        float wh = ew[e] * rinv[src[e]] * rinv[dst[e]];
        w_hat[e] = wh;
        unsafeAtomicAdd(&deg[dst[e]], wh);
    }
}

__global__ void k_dinv(const float* __restrict__ deg, float* __restrict__ dinv, int N) {
    int i = blockIdx.x * blockDim.x + threadIdx.x;
    if (i < N) { float d = deg[i]; dinv[i] = (d > 0.0f) ? rsqrtf(d) : 0.0f; }
}

__global__ void k_norm(const int* __restrict__ src, const int* __restrict__ dst,
                       const float* __restrict__ dinv, const float* __restrict__ w_hat,
                       float* __restrict__ norm, int E) {
    int e = blockIdx.x * blockDim.x + threadIdx.x;
    if (e < E) norm[e] = dinv[src[e]] * w_hat[e] * dinv[dst[e]];
}

// ------------------- counting sort of edges by destination ------------------

__global__ void k_count(const int* __restrict__ dst, int* __restrict__ cnt, int E) {
    int e = blockIdx.x * blockDim.x + threadIdx.x;
    if (e < E) atomicAdd(&cnt[dst[e]], 1);
}

// Block-local exclusive scan (256 elems/block) + per-block totals.
__global__ void __launch_bounds__(256)
k_scan1(const int* __restrict__ cnt, int* __restrict__ row_start,
        int* __restrict__ bsum, int N) {
    __shared__ int s[256];
    const int t = threadIdx.x;
    const int i = blockIdx.x * 256 + t;
    const int v = (i < N) ? cnt[i] : 0;
    s[t] = v;
    __syncthreads();
#pragma unroll
    for (int off = 1; off < 256; off <<= 1) {
        int add = (t >= off) ? s[t - off] : 0;
        __syncthreads();
        s[t] += add;
        __syncthreads();
    }
    if (i < N) row_start[i] = s[t] - v;          // exclusive within block
    if (t == 255) bsum[blockIdx.x] = s[255];     // block total
}

// Single-block exclusive scan of the (<=256) block totals, in place.
__global__ void __launch_bounds__(256)
k_scan2(int* __restrict__ bsum, int nb) {
    __shared__ int s[256];
    const int t = threadIdx.x;
    const int v = (t < nb) ? bsum[t] : 0;
    s[t] = v;
    __syncthreads();
#pragma unroll
    for (int off = 1; off < 256; off <<= 1) {
        int add = (t >= off) ? s[t - off] : 0;
        __syncthreads();
        s[t] += add;
        __syncthreads();
    }
    if (t < nb) bsum[t] = s[t] - v;              // exclusive
}

__global__ void k_scan3(int* __restrict__ row_start, int* __restrict__ cursor,
                        const int* __restrict__ bsum, int N, int E) {
    int i = blockIdx.x * blockDim.x + threadIdx.x;
    if (i < N) {
        int r = row_start[i] + bsum[blockIdx.x * 256 / 256 == 0 ? blockIdx.x : blockIdx.x];
        r = row_start[i] + bsum[blockIdx.x];
        row_start[i] = r;
        cursor[i]    = r;
    }
    if (i == 0) row_start[N] = E;
}

__global__ void k_scatter(const int* __restrict__ dst, int* __restrict__ cursor,
                          int* __restrict__ perm, int E) {
    int e = blockIdx.x * blockDim.x + threadIdx.x;
    if (e < E) {
        int p = atomicAdd(&cursor[dst[e]], 1);
        perm[p] = e;
    }
}

// x0 = feat / rowsum(feat).  One wave32 per node; lane owns a float4.
__global__ void k_rownorm(const float* __restrict__ feat, float* __restrict__ X, int N) {
    int g = blockIdx.x * blockDim.x + threadIdx.x;
    int node = g >> 5;
    if (node >= N) return;
    int lane = g & 31;
    const float4 v = ((const float4*)(feat + (size_t)node * 128))[lane];
    float s = v.x + v.y + v.z + v.w;
#pragma unroll
    for (int off = 16; off > 0; off >>= 1) s += __shfl_xor(s, off, 32);
    float inv = 1.0f / s;
    float4 o; o.x = v.x * inv; o.y = v.y * inv; o.z = v.z * inv; o.w = v.w * inv;
    ((float4*)(X + (size_t)node * 128))[lane] = o;
}

// ------------------------------ WMMA GEMM ----------------------------------
// Y[N,128] = X[N,128] @ W[128,128].  Block = 256 threads = 8 waves.
// Block covers 16 rows x 128 cols; each wave owns one 16x16 tile.
// A panel (16x128 fp32) staged into LDS by CDNA5 async loads (ASYNCcnt).

__global__ void __launch_bounds__(256)
k_gemm_wmma(const float* __restrict__ X, const float* __restrict__ W,
            float* __restrict__ Y, int N) {
    __shared__ float As[16 * 132];               // +4 pad: lane stride 132 -> distinct banks
    const int row0 = blockIdx.x * 16;
    const int tid  = threadIdx.x;

    // Async fill: 512 float4s, 256 lanes x 2 issues of b128.
#pragma unroll
    for (int p = 0; p < 2; ++p) {
        int f  = tid + p * 256;                  // float4 index in panel
        int r  = f >> 5;                         // 32 float4 per row
        int c4 = (f & 31) << 2;                  // float column (multiple of 4)
        int rg = row0 + r; if (rg > N - 1) rg = N - 1;   // clamp, keep EXEC full
        unsigned long long ga =
            (unsigned long long)(uintptr_t)(X + (size_t)rg * 128 + c4);
        // LDS byte offset = low 32 bits of the generic pointer (aperture-based).
        unsigned lo = (unsigned)(uintptr_t)(&As[r * 132 + c4]);
        asm volatile("global_load_async_to_lds_b128 %0, %1, off"
                     :: "v"(lo), "v"(ga) : "memory");
    }
    asm volatile("s_wait_asynccnt 0" ::: "memory");
    __syncthreads();

    const int wave = tid >> 5;
    const int lane = tid & 31;
    const int half = lane >> 4;      // 0: lanes 0-15, 1: lanes 16-31
    const int l    = lane & 15;
    const int col0 = wave * 16;

    v8f acc = {};
#pragma unroll
    for (int k = 0; k < 128; k += 4) {
        const int kk = k + 2 * half;             // ISA 32-bit A/B fragment layout
        v2f a, b;
        a.x = As[l * 132 + kk];
        a.y = As[l * 132 + kk + 1];
        b.x = W[(size_t)kk * 128 + col0 + l];
        b.y = W[(size_t)(kk + 1) * 128 + col0 + l];
        acc = wmma_f32_16x16x4(a, b, acc);
    }

    // C/D layout: VGPR i -> M = i + 8*half, N = l.
#pragma unroll
    for (int i = 0; i < 8; ++i) {
        int m = row0 + i + 8 * half;
        if (m < N) Y[(size_t)m * 128 + col0 + l] = acc[i];
    }
}

// --------------------- pull-mode aggregation (no atomics) -------------------
// out[n,:] = act( sum_{e in in(n)} norm[e] * Y[src[e],:]  + bias )
// One wave32 per node; lane owns 4 consecutive features.

__global__ void k_aggregate(const float* __restrict__ Y, const int* __restrict__ src,
                            const float* __restrict__ norm,
                            const int* __restrict__ row_start,
                            const int* __restrict__ perm,
                            const float* __restrict__ bias,
                            float* __restrict__ out, int N, int leaky) {
    int g = blockIdx.x * blockDim.x + threadIdx.x;
    int node = g >> 5;
    if (node >= N) return;
    int lane = g & 31;
    const int beg = row_start[node];
    const int end = row_start[node + 1];
    float4 acc; acc.x = acc.y = acc.z = acc.w = 0.0f;
    for (int idx = beg; idx < end; ++idx) {
        const int   e = perm[idx];
        const float s = norm[e];
        const float4 v = ((const float4*)(Y + (size_t)src[e] * 128))[lane];
        acc.x += s * v.x; acc.y += s * v.y; acc.z += s * v.z; acc.w += s * v.w;
    }
    const float4 b4 = ((const float4*)bias)[lane];
    acc.x += b4.x; acc.y += b4.y; acc.z += b4.z; acc.w += b4.w;
    if (leaky) {
        acc.x = (acc.x > 0.0f) ? acc.x : LEAKY_SLOPE * acc.x;
        acc.y = (acc.y > 0.0f) ? acc.y : LEAKY_SLOPE * acc.y;
        acc.z = (acc.z > 0.0f) ? acc.z : LEAKY_SLOPE * acc.z;
        acc.w = (acc.w > 0.0f) ? acc.w : LEAKY_SLOPE * acc.w;
    }
    ((float4*)(out + (size_t)node * 128))[lane] = acc;
}

// ------------------------------- launcher ----------------------------------

extern "C" void kernel_launch(void* const* d_in, const int* in_sizes, int n_in,
                              void* d_out, int out_size, void* d_ws, size_t ws_size,
                              hipStream_t stream) {
    // setup_inputs order: feat, edge_index, edge_weight, W1,b1, W2,b2, W3,b3
    const float* feat = (const float*)d_in[0];
    const int*   ei   = (const int*)d_in[1];   // [2, E] flat: row0=src, row1=dst
    const float* ew   = (const float*)d_in[2];
    const float* W1   = (const float*)d_in[3];
    const float* b1   = (const float*)d_in[4];
    const float* W2   = (const float*)d_in[5];
    const float* b2   = (const float*)d_in[6];
    const float* W3   = (const float*)d_in[7];
    const float* b3   = (const float*)d_in[8];

    const int E = in_sizes[2];
    const int N = in_sizes[0] / 128;
    const int* src = ei;
    const int* dst = ei + E;

    float* ws     = (float*)d_ws;
    float* X      = ws;                          // N*128  layer input
    float* Y      = X + (size_t)N * 128;         // N*128  X @ W
    float* rowsum = Y + (size_t)N * 128;         // N
    float* rinv   = rowsum + N;                  // N
    float* deg    = rinv + N;                    // N
    float* dinv   = deg + N;                     // N
    float* w_hat  = dinv + N;                    // E
    float* nrm    = w_hat + E;                   // E
    int*   cnt      = (int*)(nrm + E);           // N
    int*   row_start= cnt + N;                   // N+1
    int*   cursor   = row_start + (N + 1);       // N
    int*   bsum     = cursor + N;                // <=256
    int*   perm     = bsum + 256;                // E
    float* out    = (float*)d_out;

    const int TPB = 256;
    const int gN  = (N + TPB - 1) / TPB;         // also #scan blocks
    const int gE  = (E + TPB - 1) / TPB;
    const int gNW = ((N * 32) + TPB - 1) / TPB;  // wave-per-node
    const int gG  = (N + 15) / 16;               // GEMM row panels

    // --- normalization coefficients (shared by all layers) ---
    k_clear    <<<gN, TPB, 0, stream>>>(rowsum, deg, cnt, N);
    k_rowsum   <<<gE, TPB, 0, stream>>>(src, ew, rowsum, E);
    k_rinv     <<<gN, TPB, 0, stream>>>(rowsum, rinv, N);
    k_what_deg <<<gE, TPB, 0, stream>>>(src, dst, ew, rinv, w_hat, deg, E);
    k_dinv     <<<gN, TPB, 0, stream>>>(deg, dinv, N);
    k_norm     <<<gE, TPB, 0, stream>>>(src, dst, dinv, w_hat, nrm, E);

    // --- counting sort of edges by destination (shared by all layers) ---
    k_count    <<<gE, TPB, 0, stream>>>(dst, cnt, E);
    k_scan1    <<<gN, TPB, 0, stream>>>(cnt, row_start, bsum, N);
    k_scan2    <<<1,  TPB, 0, stream>>>(bsum, gN);
    k_scan3    <<<gN, TPB, 0, stream>>>(row_start, cursor, bsum, N, E);
    k_scatter  <<<gE, TPB, 0, stream>>>(dst, cursor, perm, E);

    // --- input row-normalization ---
    k_rownorm  <<<gNW, TPB, 0, stream>>>(feat, X, N);

    // --- layer 1 ---
    k_gemm_wmma<<<gG,  TPB, 0, stream>>>(X, W1, Y, N);
    k_aggregate<<<gNW, TPB, 0, stream>>>(Y, src, nrm, row_start, perm, b1, X, N, 1);
    // --- layer 2 ---
    k_gemm_wmma<<<gG,  TPB, 0, stream>>>(X, W2, Y, N);
    k_aggregate<<<gNW, TPB, 0, stream>>>(Y, src, nrm, row_start, perm, b2, X, N, 1);
    // --- layer 3 (no activation, straight into d_out) ---
    k_gemm_wmma<<<gG,  TPB, 0, stream>>>(X, W3, Y, N);
    k_aggregate<<<gNW, TPB, 0, stream>>>(Y, src, nrm, row_start, perm, b3, out, N, 0);
}